// RBF_4647154614792
// MI455X (gfx1250) — compile-verified
//
#include <hip/hip_runtime.h>

// RBF layer via  d^2 = |x|^2 + |c|^2 - 2 x.c  ->  the cross term is a
// 1024x1024x256 fp32 GEMM mapped onto V_WMMA_F32_16X16X4_F32 (CDNA5).
// Problem is memory-bound (~6 MB traffic vs 0.54 GFLOP), so full-fp32
// matrix math is free and matches the reference numerics.

typedef __attribute__((ext_vector_type(2))) float v2f;
typedef __attribute__((ext_vector_type(4))) float v4f;
typedef __attribute__((ext_vector_type(8))) float v8f;

#define NDIM 1024   // rows of x
#define ODIM 1024   // centres (output features)
#define KDIM 256    // in_features
#define BM   64     // block tile rows
#define BN   128    // block tile cols
#define KC   64     // K chunk staged in LDS
#define LDT  68     // padded LDS row pitch (floats): bank-conflict-free column reads

__global__ __launch_bounds__(256) void rbf_wmma_f32_kernel(
    const float* __restrict__ x, const float* __restrict__ centres,
    const float* __restrict__ log_sigmas, float* __restrict__ out)
{
  __shared__ float xs[BM * LDT];   // X tile   64 x KC
  __shared__ float cs[BN * LDT];   // C tile  128 x KC
  __shared__ float xn[BM];         // |x_row|^2
  __shared__ float cn[BN];         // |c_row|^2

  const int tid  = threadIdx.x;
  const int lane = tid & 31;        // wave32
  const int wave = tid >> 5;        // 8 waves
  const int lm   = lane & 15;       // M/N index within a 16-wide fragment
  const int h    = lane >> 4;       // lane half selects K pair (ISA A/B layout)

  const int gn0 = blockIdx.x * BN;  // centre block origin
  const int gm0 = blockIdx.y * BM;  // x-row block origin

  // ---- squared-norm pre-pass (L2-hot, float4 loads) ----
  if (tid < BM) {
    const v4f* p = (const v4f*)(x + (size_t)(gm0 + tid) * KDIM);
    float s = 0.f;
    #pragma unroll 8
    for (int k = 0; k < KDIM / 4; ++k) {
      v4f v = p[k];
      s += v.x * v.x + v.y * v.y + v.z * v.z + v.w * v.w;
    }
    xn[tid] = s;
  } else if (tid < BM + BN) {
    const v4f* p = (const v4f*)(centres + (size_t)(gn0 + (tid - BM)) * KDIM);
    float s = 0.f;
    #pragma unroll 8
    for (int k = 0; k < KDIM / 4; ++k) {
      v4f v = p[k];
      s += v.x * v.x + v.y * v.y + v.z * v.z + v.w * v.w;
    }
    cn[tid - BM] = s;
  }

  // wave -> 32x32 sub-tile of the 64x128 block tile
  const int wm = (wave & 1) * 32;   // 2 waves along M
  const int wn = (wave >> 1) * 32;  // 4 waves along N

  v8f acc[2][2] = {};               // 2x2 WMMA accumulators (fp32)

  for (int kc = 0; kc < KDIM; kc += KC) {
    // ---- stage X tile: 64x64 floats = 1024 v4f, 4 per thread ----
    #pragma unroll
    for (int p = 0; p < 4; ++p) {
      int q  = tid + p * 256;
      int r  = q >> 4;              // row 0..63
      int kv = (q & 15) << 2;       // k 0..60 step 4
      *(v4f*)&xs[r * LDT + kv] =
          *(const v4f*)(x + (size_t)(gm0 + r) * KDIM + kc + kv);
    }
    // ---- stage C tile: 128x64 floats = 2048 v4f, 8 per thread ----
    #pragma unroll
    for (int p = 0; p < 8; ++p) {
      int q  = tid + p * 256;
      int r  = q >> 4;              // row 0..127
      int kv = (q & 15) << 2;
      *(v4f*)&cs[r * LDT + kv] =
          *(const v4f*)(centres + (size_t)(gn0 + r) * KDIM + kc + kv);
    }
    __syncthreads();

    // warm next K chunk in cache while WMMAs run (global_prefetch_b8)
    if (kc + KC < KDIM) {
      __builtin_prefetch(x + (size_t)(gm0 + (tid >> 2)) * KDIM + kc + KC, 0, 1);
      __builtin_prefetch(centres + (size_t)(gn0 + (tid >> 1)) * KDIM + kc + KC, 0, 1);
    }

    // ---- 16 k-steps of 4, 4 WMMAs each ----
    #pragma unroll
    for (int k0 = 0; k0 < KC; k0 += 4) {
      const int kk = k0 + 2 * h;    // lane-half K offset (A/B VGPR layout)
      v2f a0 = *(const v2f*)&xs[(wm +      lm) * LDT + kk];
      v2f a1 = *(const v2f*)&xs[(wm + 16 + lm) * LDT + kk];
      v2f b0 = *(const v2f*)&cs[(wn +      lm) * LDT + kk];
      v2f b1 = *(const v2f*)&cs[(wn + 16 + lm) * LDT + kk];
      acc[0][0] = __builtin_amdgcn_wmma_f32_16x16x4_f32(
          false, a0, false, b0, (short)0, acc[0][0], false, false);
      acc[0][1] = __builtin_amdgcn_wmma_f32_16x16x4_f32(
          false, a0, false, b1, (short)0, acc[0][1], false, false);
      acc[1][0] = __builtin_amdgcn_wmma_f32_16x16x4_f32(
          false, a1, false, b0, (short)0, acc[1][0], false, false);
      acc[1][1] = __builtin_amdgcn_wmma_f32_16x16x4_f32(
          false, a1, false, b1, (short)0, acc[1][1], false, false);
    }
    __syncthreads();
  }

  // ---- epilogue: d2 = xn + cn - 2S ; out = exp(-0.5 * d2 * exp(-2*ls)) ----
  #pragma unroll
  for (int j = 0; j < 2; ++j) {
    const int nb = wn + j * 16 + lm;          // column within block tile
    const int gc = gn0 + nb;                  // global centre index
    const float invs2 = __expf(-2.0f * log_sigmas[gc]);
    const float cno = cn[nb];
    #pragma unroll
    for (int i = 0; i < 2; ++i) {
      const int rbase = wm + i * 16 + 8 * h;  // C/D layout: row = e + 8h
      #pragma unroll
      for (int e = 0; e < 8; ++e) {
        const int rb = rbase + e;
        const float s  = acc[i][j][e];
        const float d2 = xn[rb] + cno - 2.0f * s;
        out[(size_t)(gm0 + rb) * ODIM + gc] = __expf(-0.5f * d2 * invs2);
      }
    }
  }
}

extern "C" void kernel_launch(void* const* d_in, const int* in_sizes, int n_in,
                              void* d_out, int out_size, void* d_ws, size_t ws_size,
                              hipStream_t stream) {
  const float* x       = (const float*)d_in[0];
  const float* centres = (const float*)d_in[1];
  const float* lsig    = (const float*)d_in[2];
  float* out           = (float*)d_out;

  dim3 grid(ODIM / BN, NDIM / BM);   // 8 x 16 = 128 workgroups
  rbf_wmma_f32_kernel<<<grid, 256, 0, stream>>>(x, centres, lsig, out);
}